// HausdorffERLoss_22325240004965
// MI455X (gfx1250) — compile-verified
//
#include <hip/hip_runtime.h>
#include <hip/hip_bf16.h>

typedef __attribute__((ext_vector_type(2))) float v2f;
typedef __attribute__((ext_vector_type(8))) float v8f;

#define IMG_B   16
#define IMG_H   1024
#define IMG_W   1024
#define IMG_N   (IMG_H * IMG_W)          // 1<<20 per sample
#define EROSIONS 10

// Stencil tiling: workgroup = 128 threads (4 waves), each wave owns a 16x16 tile,
// WG covers 16 rows x 64 cols. Staged LDS region = 18 x 66 (1-halo), pitch 67.
#define TS_ROWS  18
#define TS_COLS  66
#define TS_PITCH 67
#define TILES_X  (IMG_W / 64)            // 16
#define TILES_Y  (IMG_H / 16)            // 64
#define TILES_PER_IMG (TILES_X * TILES_Y) // 1024

// ---------------------------------------------------------------------------
// Kernel 0: bound = (pred - target)^2, vectorized float4
// ---------------------------------------------------------------------------
__global__ __launch_bounds__(256) void bound_kernel(const float* __restrict__ p,
                                                    const float* __restrict__ t,
                                                    float* __restrict__ o, int n4) {
  int i = blockIdx.x * blockDim.x + threadIdx.x;
  if (i < n4) {
    float4 a = ((const float4*)p)[i];
    float4 b = ((const float4*)t)[i];
    float4 r;
    r.x = (a.x - b.x) * (a.x - b.x);
    r.y = (a.y - b.y) * (a.y - b.y);
    r.z = (a.z - b.z) * (a.z - b.z);
    r.w = (a.w - b.w) * (a.w - b.w);
    ((float4*)o)[i] = r;
  }
}

// ---------------------------------------------------------------------------
// Per-iteration stencil kernel.
//   in      : previous raw ero (or bound for iter 0), normalized on the fly
//   out     : this iteration's raw ero = max(0.2*cross5 - 0.5, 0)
//   mnPrev/mxPrev : per-sample min/max of previous iteration (nullptr => skip)
//   psum/pmin/pmax: per-workgroup partials, unique slot per WG (deterministic)
// Cross stencil via WMMA band-matrix trick: cross5 = T*X + X*T - X.
// ---------------------------------------------------------------------------
__global__ __launch_bounds__(128) void stencil_kernel(
    const float* __restrict__ in, float* __restrict__ out,
    const float* __restrict__ mnPrev, const float* __restrict__ mxPrev,
    float* __restrict__ psum, float* __restrict__ pmin, float* __restrict__ pmax) {
  __shared__ float tile[TS_ROWS * TS_PITCH];
  __shared__ float redS[4], redMn[4], redMx[4];

  const int b  = blockIdx.z;
  const int y0 = blockIdx.y << 4;   // 16-row band
  const int x0 = blockIdx.x << 6;   // 64-col band
  const int tid = threadIdx.x;
  const size_t imgOff = (size_t)b << 20;
  const float* src = in + imgOff;

  // Normalization of the previous iteration's output, applied at load.
  float mn = 0.f, inv = 1.f;
  int doNorm = 0;
  if (mnPrev) {
    mn = mnPrev[b];
    float ptp = mxPrev[b] - mn;
    if (ptp != 0.f) { doNorm = 1; inv = 1.f / ptp; }
  }

  // Cooperative staging of 18x66 region (zero padding outside image).
  for (int i = tid; i < TS_ROWS * TS_COLS; i += 128) {
    int r = i / TS_COLS, c = i - r * TS_COLS;
    int gy = y0 - 1 + r, gx = x0 - 1 + c;
    float v = 0.f;
    if ((unsigned)gy < (unsigned)IMG_H && (unsigned)gx < (unsigned)IMG_W) {
      v = src[((size_t)gy << 10) + gx];
      if (doNorm) v = (v - mn) * inv;
    }
    tile[r * TS_PITCH + c] = v;
  }
  __syncthreads();

  const int wave  = tid >> 5;       // 0..3 : which 16-col tile
  const int lane  = tid & 31;
  const int h     = lane >> 4;      // half-wave
  const int nn    = lane & 15;
  const int cbase = 1 + (wave << 4);// staged col of tile col 0

  // Tridiagonal-ones T in the (symmetric) A/B chunk layout:
  // chunk c, element j: T[nn][4c + j + 2h]
  v2f t[4];
#pragma unroll
  for (int c = 0; c < 4; ++c) {
    int k0 = 4 * c + 2 * h;
    int d0 = nn - k0, d1 = nn - (k0 + 1);
    t[c].x = (d0 >= -1 && d0 <= 1) ? 1.f : 0.f;
    t[c].y = (d1 >= -1 && d1 <= 1) ? 1.f : 0.f;
  }

  v8f acc = {};
  // acc = T * X   (vertical 3-sum): B chunk c elem j = X[4c+j+2h][nn]
#pragma unroll
  for (int c = 0; c < 4; ++c) {
    int k0 = 4 * c + 2 * h;
    v2f xb = { tile[(k0 + 1) * TS_PITCH + cbase + nn],
               tile[(k0 + 2) * TS_PITCH + cbase + nn] };
    acc = __builtin_amdgcn_wmma_f32_16x16x4_f32(false, t[c], false, xb,
                                                (short)0, acc, false, false);
  }
  // acc += X * T  (horizontal 3-sum): A chunk c elem j = X[nn][4c+j+2h]
#pragma unroll
  for (int c = 0; c < 4; ++c) {
    int k0 = 4 * c + 2 * h;
    v2f xa = { tile[(nn + 1) * TS_PITCH + cbase + k0],
               tile[(nn + 1) * TS_PITCH + cbase + k0 + 1] };
    acc = __builtin_amdgcn_wmma_f32_16x16x4_f32(false, xa, false, t[c],
                                                (short)0, acc, false, false);
  }

  // Vertical halo fix: row 0 (+up neighbor), row 15 (+down neighbor).
  // D layout: acc[r] holds row r (lanes 0-15) / row r+8 (lanes 16-31), col nn.
  {
    float topv = tile[0 * TS_PITCH + cbase + nn];
    float botv = tile[17 * TS_PITCH + cbase + nn];
    if (h == 0) acc[0] += topv; else acc[7] += botv;
  }

  // Horizontal halo + center subtract + clamp + store + local reduce.
  float s_  = 0.f;
  float mns = 3.4e38f;
  float mxs = -3.4e38f;
  float* dst = out + imgOff + ((size_t)y0 << 10) + (x0 + (wave << 4));
#pragma unroll
  for (int r = 0; r < 8; ++r) {
    int row  = r + 8 * h;
    int srow = row + 1;
    float hv = 0.f;
    if (nn == 0)  hv = tile[srow * TS_PITCH + cbase - 1];
    if (nn == 15) hv = tile[srow * TS_PITCH + cbase + 16];
    float center = tile[srow * TS_PITCH + cbase + nn];
    float sum5 = acc[r] + hv - center;           // cross5 = T*X + X*T - X (+halos)
    float ero  = fmaxf(0.2f * sum5 - 0.5f, 0.f);
    dst[((size_t)row << 10) + nn] = ero;
    s_  += ero;
    mns  = fminf(mns, ero);
    mxs  = fmaxf(mxs, ero);
  }

  // Wave32 butterfly reduction (deterministic).
#pragma unroll
  for (int m = 16; m >= 1; m >>= 1) {
    s_  += __shfl_xor(s_, m, 32);
    mns  = fminf(mns, __shfl_xor(mns, m, 32));
    mxs  = fmaxf(mxs, __shfl_xor(mxs, m, 32));
  }
  if (lane == 0) { redS[wave] = s_; redMn[wave] = mns; redMx[wave] = mxs; }
  __syncthreads();
  if (tid == 0) {
    float S  = redS[0] + redS[1] + redS[2] + redS[3];
    float Mn = fminf(fminf(redMn[0], redMn[1]), fminf(redMn[2], redMn[3]));
    float Mx = fmaxf(fmaxf(redMx[0], redMx[1]), fmaxf(redMx[2], redMx[3]));
    int slot = b * TILES_PER_IMG + blockIdx.y * TILES_X + blockIdx.x;
    psum[slot] = S; pmin[slot] = Mn; pmax[slot] = Mx;
  }
}

// ---------------------------------------------------------------------------
// Per-iteration min/max partial reduction (needed before next stencil).
// One block per sample; deterministic tree reduce over 1024 partials.
// ---------------------------------------------------------------------------
__global__ __launch_bounds__(256) void minmax_reduce(const float* __restrict__ pmin,
                                                     const float* __restrict__ pmax,
                                                     float* __restrict__ mnOut,
                                                     float* __restrict__ mxOut) {
  __shared__ float smn[256], smx[256];
  int b = blockIdx.x, tid = threadIdx.x;
  const float* pm = pmin + b * TILES_PER_IMG;
  const float* px = pmax + b * TILES_PER_IMG;
  float mn = 3.4e38f, mx = -3.4e38f;
  for (int i = tid; i < TILES_PER_IMG; i += 256) {
    mn = fminf(mn, pm[i]);
    mx = fmaxf(mx, px[i]);
  }
  smn[tid] = mn; smx[tid] = mx;
  __syncthreads();
  for (int off = 128; off; off >>= 1) {
    if (tid < off) {
      smn[tid] = fminf(smn[tid], smn[tid + off]);
      smx[tid] = fmaxf(smx[tid], smx[tid + off]);
    }
    __syncthreads();
  }
  if (tid == 0) { mnOut[b] = smn[0]; mxOut[b] = smx[0]; }
}

// ---------------------------------------------------------------------------
// Final scalar: mean of per-iteration/per-sample normalized sums.
// mean(norm(ero)) = (S - N*mn)/ptp if ptp != 0 else S, divided by B*N overall.
// ---------------------------------------------------------------------------
__global__ __launch_bounds__(256) void final_reduce(const float* __restrict__ psum,
                                                    const float* __restrict__ mnArr,
                                                    const float* __restrict__ mxArr,
                                                    float* __restrict__ outp) {
  __shared__ float ss[256];
  __shared__ float totalSh;
  int tid = threadIdx.x;
  if (tid == 0) totalSh = 0.f;
  __syncthreads();
  for (int k = 0; k < EROSIONS; ++k) {
    float w = (float)((k + 1) * (k + 1));
    for (int b = 0; b < IMG_B; ++b) {
      const float* p = psum + (size_t)(k * IMG_B + b) * TILES_PER_IMG;
      float s = 0.f;
      for (int i = tid; i < TILES_PER_IMG; i += 256) s += p[i];
      ss[tid] = s;
      __syncthreads();
      for (int off = 128; off; off >>= 1) {
        if (tid < off) ss[tid] += ss[tid + off];
        __syncthreads();
      }
      if (tid == 0) {
        float S   = ss[0];
        float mn  = mnArr[k * IMG_B + b];
        float ptp = mxArr[k * IMG_B + b] - mn;
        float Sn  = (ptp != 0.f) ? (S - (float)IMG_N * mn) / ptp : S;
        totalSh += w * Sn;
      }
      __syncthreads();
    }
  }
  if (tid == 0) outp[0] = totalSh / ((float)IMG_B * (float)IMG_N);
}

// ---------------------------------------------------------------------------
// Host-side orchestration. All launches on `stream`; graph-capture safe.
// Workspace map (floats): [bufA 16M][bufB 16M][psum 160K][pmin 160K][pmax 160K]
//                         [mn 160][mx 160]   (~130 MB total)
// ---------------------------------------------------------------------------
extern "C" void kernel_launch(void* const* d_in, const int* in_sizes, int n_in,
                              void* d_out, int out_size, void* d_ws, size_t ws_size,
                              hipStream_t stream) {
  (void)in_sizes; (void)n_in; (void)out_size; (void)ws_size;
  const float* pred   = (const float*)d_in[0];
  const float* target = (const float*)d_in[1];
  float* ws = (float*)d_ws;

  const size_t NTOT = (size_t)IMG_B * IMG_N;          // 16M elements
  float* bufA = ws;
  float* bufB = ws + NTOT;
  float* psum = ws + 2 * NTOT;                        // 10*16*1024 partials
  float* pmin = psum + (size_t)EROSIONS * IMG_B * TILES_PER_IMG;
  float* pmax = pmin + (size_t)EROSIONS * IMG_B * TILES_PER_IMG;
  float* mnA  = pmax + (size_t)EROSIONS * IMG_B * TILES_PER_IMG;
  float* mxA  = mnA + EROSIONS * IMG_B;

  // bound = (pred - target)^2  -> bufA
  {
    int n4 = (int)(NTOT / 4);
    bound_kernel<<<(n4 + 255) / 256, 256, 0, stream>>>(pred, target, bufA, n4);
  }

  dim3 grid(TILES_X, TILES_Y, IMG_B);
  for (int k = 0; k < EROSIONS; ++k) {
    const float* in  = (k & 1) ? bufB : bufA;
    float*       out = (k & 1) ? bufA : bufB;
    const float* mnPrev = (k > 0) ? (mnA + (k - 1) * IMG_B) : nullptr;
    const float* mxPrev = (k > 0) ? (mxA + (k - 1) * IMG_B) : nullptr;
    size_t po = (size_t)k * IMG_B * TILES_PER_IMG;
    stencil_kernel<<<grid, 128, 0, stream>>>(in, out, mnPrev, mxPrev,
                                             psum + po, pmin + po, pmax + po);
    minmax_reduce<<<IMG_B, 256, 0, stream>>>(pmin + po, pmax + po,
                                             mnA + k * IMG_B, mxA + k * IMG_B);
  }

  final_reduce<<<1, 256, 0, stream>>>(psum, mnA, mxA, (float*)d_out);
}